// ObjectDetection_7885559955839
// MI455X (gfx1250) — compile-verified
//
#include <hip/hip_runtime.h>

#define NPROP   8192
#define CLASSES 81
#define PAD     300
#define NMS_THR 0.5f
#define CHUNK   8                                   // classes per LDS tile
#define NCHUNK  ((CLASSES + CHUNK - 1) / CHUNK)     // 11 (10 full + 1 of size 1)

// ---------------------------------------------------------------------------
// CDNA5 async-copy helpers (gfx1250): per-lane DMA global -> LDS, tracked by
// ASYNCcnt.  LDS address comes from a VGPR; global address is a 64-bit SGPR base plus a VGPR offset.
// A __shared__ pointer's low 32 bits are exactly the LDS byte offset.
// ---------------------------------------------------------------------------
__device__ __forceinline__ unsigned int lds_off(const void* p) {
    return (unsigned int)(unsigned long long)p;
}
__device__ __forceinline__ void async_ld_b128(unsigned int ldsaddr,
                                              unsigned int voff,
                                              const void* base) {
    asm volatile("global_load_async_to_lds_b128 %0, %1, %2"
                 :: "v"(ldsaddr), "v"(voff),
                    "s"((unsigned long long)base)
                 : "memory");
}
__device__ __forceinline__ void async_ld_b32(unsigned int ldsaddr,
                                             unsigned int voff,
                                             const void* base) {
    asm volatile("global_load_async_to_lds_b32 %0, %1, %2"
                 :: "v"(ldsaddr), "v"(voff),
                    "s"((unsigned long long)base)
                 : "memory");
}
// Partial waits: async loads complete in order, so waiting "<= N" after
// issuing N ops for the NEXT chunk guarantees the CURRENT chunk has landed.
#define WAIT_ASYNC_0()  asm volatile("s_wait_asynccnt 0x0" ::: "memory")
#define WAIT_ASYNC_2()  asm volatile("s_wait_asynccnt 0x2" ::: "memory")
#define WAIT_ASYNC_16() asm volatile("s_wait_asynccnt 0x10" ::: "memory")

// ---------------------------------------------------------------------------
// Kernel 1: bbox decode + per-row class argmax.  One thread per proposal,
// 256 proposals per block.  deltas/scores staged into double-buffered LDS
// tiles with async DMA; DMA for chunk i+1 overlaps consumption of chunk i.
// ---------------------------------------------------------------------------
__shared__ __align__(16) float g_tile_d[2][256 * CHUNK * 4];  // 64 KB
__shared__ __align__(16) float g_tile_s[2][256 * CHUNK];      // 16 KB

__device__ __forceinline__ void issue_chunk(int ci, int buf, int p0, int tid,
                                            const float* __restrict__ deltas,
                                            const float* __restrict__ scores)
{
    const int c0 = ci * CHUNK;
    const int nc = (CLASSES - c0 < CHUNK) ? (CLASSES - c0) : CHUNK;
    // deltas tile: per proposal row, nc 16-byte groups (4 floats per class).
    // Row stride in memory is 324 floats = 1296 B = 81*16 B -> 16B aligned.
    for (int l = tid; l < 256 * nc; l += 256) {
        const int pp = l / nc, g = l % nc;
        const unsigned int voff =
            (unsigned int)((((p0 + pp) * (CLASSES * 4)) + (c0 + g) * 4) * 4);
        async_ld_b128(lds_off(&g_tile_d[buf][(pp * nc + g) * 4]), voff, deltas);
    }
    // scores tile: one f32 per (proposal, class)
    for (int l = tid; l < 256 * nc; l += 256) {
        const int pp = l / nc, c = l % nc;
        const unsigned int voff =
            (unsigned int)((((p0 + pp) * CLASSES) + c0 + c) * 4);
        async_ld_b32(lds_off(&g_tile_s[buf][pp * nc + c]), voff, scores);
    }
}

__global__ __launch_bounds__(256)
void decode_kernel(const float* __restrict__ meta,
                   const float* __restrict__ deltas,
                   const float* __restrict__ proposals,
                   const float* __restrict__ scores,
                   float4* __restrict__ selBoxes,
                   float* __restrict__ maxScores)
{
    const int tid = threadIdx.x;
    const int p0  = blockIdx.x * 256;
    const int p   = p0 + tid;

    const float Himg  = meta[0];
    const float Wimg  = meta[1];
    const float scale = meta[2];

    const float4 pr = reinterpret_cast<const float4*>(proposals)[p];
    const float x1 = pr.x / scale, y1 = pr.y / scale;
    const float x2 = pr.z / scale, y2 = pr.w / scale;
    const float w  = x2 - x1 + 1.0f, h = y2 - y1 + 1.0f;
    const float cx = x1 + 0.5f * w,  cy = y1 + 0.5f * h;

    float bestAll = -__builtin_inff();
    float bestFg  = -__builtin_inff();
    float d0 = 0.f, d1 = 0.f, d2 = 0.f, d3 = 0.f;

    // software pipeline prologue: chunk 0 -> buffer 0
    issue_chunk(0, 0, p0, tid, deltas, scores);

    for (int ci = 0; ci < NCHUNK; ++ci) {
        const int buf = ci & 1;
        const int c0  = ci * CHUNK;
        const int nc  = (CLASSES - c0 < CHUNK) ? (CLASSES - c0) : CHUNK;

        // issue next chunk into the other buffer (safe: that buffer was
        // consumed last iteration and sealed by the trailing barrier)
        if (ci + 1 < NCHUNK)
            issue_chunk(ci + 1, buf ^ 1, p0, tid, deltas, scores);

        // wait until chunk ci has fully landed (in-order completion):
        // each thread issued 2*nc_next ops for chunk ci+1 that may remain.
        if (ci + 1 >= NCHUNK)           { WAIT_ASYNC_0(); }
        else if (ci + 1 == NCHUNK - 1)  { WAIT_ASYNC_2(); }   // next nc == 1
        else                            { WAIT_ASYNC_16(); }  // next nc == 8
        __syncthreads();   // all waves' DMA for chunk ci visible

        for (int c = 0; c < nc; ++c) {
            const float s = g_tile_s[buf][tid * nc + c];
            const int cls = c0 + c;
            if (s > bestAll) {          // first-max semantics like jnp.argmax
                bestAll = s;
                d0 = g_tile_d[buf][(tid * nc + c) * 4 + 0];
                d1 = g_tile_d[buf][(tid * nc + c) * 4 + 1];
                d2 = g_tile_d[buf][(tid * nc + c) * 4 + 2];
                d3 = g_tile_d[buf][(tid * nc + c) * 4 + 3];
            }
            if (cls >= 1) bestFg = fmaxf(bestFg, s);
        }
        __syncthreads();   // buffer `buf` free for chunk ci+2's DMA
    }

    // decode best class box + clip
    const float pcx = d0 * w + cx;
    const float pcy = d1 * h + cy;
    const float pw  = expf(d2) * w;
    const float ph  = expf(d3) * h;
    float ox1 = pcx - 0.5f * pw, oy1 = pcy - 0.5f * ph;
    float ox2 = pcx + 0.5f * pw, oy2 = pcy + 0.5f * ph;
    const float Wl = Wimg - 1.0f, Hl = Himg - 1.0f;
    ox1 = fminf(fmaxf(ox1, 0.f), Wl);
    oy1 = fminf(fmaxf(oy1, 0.f), Hl);
    ox2 = fminf(fmaxf(ox2, 0.f), Wl);
    oy2 = fminf(fmaxf(oy2, 0.f), Hl);

    selBoxes[p]  = make_float4(ox1, oy1, ox2, oy2);
    maxScores[p] = bestFg;
}

// ---------------------------------------------------------------------------
// Kernel 2: single-block bitonic argsort (descending score, stable by index),
// 8192 (key,idx) pairs in 64 KB LDS, then gather sorted boxes.
// ---------------------------------------------------------------------------
__global__ __launch_bounds__(1024)
void sort_kernel(const float* __restrict__ maxScores,
                 const float4* __restrict__ selBoxes,
                 int* __restrict__ order,
                 float4* __restrict__ sortedBoxes)
{
    __shared__ float ks[NPROP];   // 32 KB
    __shared__ int   ki[NPROP];   // 32 KB
    const int tid = threadIdx.x;

    for (int i = tid; i < NPROP; i += 1024) { ks[i] = maxScores[i]; ki[i] = i; }
    __syncthreads();

    for (int k = 2; k <= NPROP; k <<= 1) {
        for (int j = k >> 1; j > 0; j >>= 1) {
            for (int i = tid; i < NPROP; i += 1024) {
                const int ixj = i ^ j;
                if (ixj > i) {
                    const float si = ks[i], sj = ks[ixj];
                    const int   ii = ki[i], ij = ki[ixj];
                    const bool before = (si > sj) || (si == sj && ii < ij);
                    const bool dirUp  = ((i & k) == 0);
                    if (dirUp ? !before : before) {
                        ks[i] = sj; ks[ixj] = si;
                        ki[i] = ij; ki[ixj] = ii;
                    }
                }
            }
            __syncthreads();
        }
    }

    for (int i = tid; i < NPROP; i += 1024) {
        const int oi = ki[i];
        order[i]       = oi;
        sortedBoxes[i] = selBoxes[oi];
    }
}

// ---------------------------------------------------------------------------
// Kernel 3: greedy NMS, one 1024-thread block.  Each thread owns 8 sorted
// boxes in registers; suppression mask is an LDS byte array; kept box is
// broadcast through LDS.  Early exit at PAD kept.
// ---------------------------------------------------------------------------
__global__ __launch_bounds__(1024)
void nms_kernel(const float4* __restrict__ sortedBoxes,
                int* __restrict__ keepPos,
                int* __restrict__ keptCount)
{
    __shared__ unsigned char sup[NPROP];  // 8 KB
    __shared__ float bb[5];               // broadcast x1,y1,x2,y2,area
    const int tid = threadIdx.x;

    float4 b[8];
    float  area[8];
    for (int r = 0; r < 8; ++r) {
        const int j = tid * 8 + r;
        b[r] = sortedBoxes[j];
        area[r] = fmaxf(b[r].z - b[r].x, 0.f) * fmaxf(b[r].w - b[r].y, 0.f);
        sup[j] = 0;
    }
    __syncthreads();

    int kept = 0;
    for (int i = 0; i < NPROP && kept < PAD; ++i) {
        if (sup[i]) continue;             // uniform LDS read -> uniform branch
        if (tid == 0) keepPos[kept] = i;
        if (tid == (i >> 3)) {
            const int r = i & 7;
            bb[0] = b[r].x; bb[1] = b[r].y; bb[2] = b[r].z; bb[3] = b[r].w;
            bb[4] = area[r];
        }
        __syncthreads();
        const float ix1 = bb[0], iy1 = bb[1], ix2 = bb[2], iy2 = bb[3], ia = bb[4];
        for (int r = 0; r < 8; ++r) {
            const int j = tid * 8 + r;
            if (j > i) {
                const float lx = fmaxf(ix1, b[r].x), ly = fmaxf(iy1, b[r].y);
                const float rx = fminf(ix2, b[r].z), ry = fminf(iy2, b[r].w);
                const float iw = fmaxf(rx - lx, 0.f), ih = fmaxf(ry - ly, 0.f);
                const float inter = iw * ih;
                const float iou = inter / (ia + area[r] - inter + 1e-8f);
                if (iou > NMS_THR) sup[j] = 1;
            }
        }
        ++kept;
        __syncthreads();
    }
    if (tid == 0) *keptCount = kept;
}

// ---------------------------------------------------------------------------
// Kernel 4: emit outputs.  d_out = [boxes (300,4) | scores (300,81)] flat.
// Slots >= kept are zeroed (reference's validity mask).
// ---------------------------------------------------------------------------
__global__ __launch_bounds__(256)
void output_kernel(const float4* __restrict__ sortedBoxes,
                   const float* __restrict__ scores,
                   const int* __restrict__ order,
                   const int* __restrict__ keepPos,
                   const int* __restrict__ keptCount,
                   float* __restrict__ out)
{
    const int e = blockIdx.x * blockDim.x + threadIdx.x;
    const int kept = *keptCount;
    const int nbox = PAD * 4;
    const int ntot = nbox + PAD * CLASSES;
    if (e < nbox) {
        const int s = e >> 2, k = e & 3;
        float v = 0.f;
        if (s < kept) {
            const float4 bbx = sortedBoxes[keepPos[s]];
            v = (k == 0) ? bbx.x : (k == 1) ? bbx.y : (k == 2) ? bbx.z : bbx.w;
        }
        out[e] = v;
    } else if (e < ntot) {
        const int t = e - nbox;
        const int s = t / CLASSES, c = t % CLASSES;
        float v = 0.f;
        if (s < kept) {
            const int orig = order[keepPos[s]];
            v = scores[orig * CLASSES + c];
        }
        out[e] = v;
    }
}

// ---------------------------------------------------------------------------
extern "C" void kernel_launch(void* const* d_in, const int* in_sizes, int n_in,
                              void* d_out, int out_size, void* d_ws, size_t ws_size,
                              hipStream_t stream) {
    const float* meta      = (const float*)d_in[0];   // (1,3)
    const float* deltas    = (const float*)d_in[1];   // (1,N,324)
    const float* proposals = (const float*)d_in[2];   // (1,N,4)
    const float* scores    = (const float*)d_in[3];   // (1,N,81)
    float* out = (float*)d_out;                       // 300*4 + 300*81

    char* ws = (char*)d_ws;
    float4* selBoxes    = (float4*)(ws + 0);                 // 128 KB
    float*  maxScores   = (float*) (ws + 131072);            //  32 KB
    int*    order       = (int*)   (ws + 163840);            //  32 KB
    float4* sortedBoxes = (float4*)(ws + 196608);            // 128 KB
    int*    keepPos     = (int*)   (ws + 327680);            // 1.2 KB
    int*    keptCount   = (int*)   (ws + 327680 + PAD * sizeof(int));

    decode_kernel<<<NPROP / 256, 256, 0, stream>>>(meta, deltas, proposals,
                                                   scores, selBoxes, maxScores);
    sort_kernel<<<1, 1024, 0, stream>>>(maxScores, selBoxes, order, sortedBoxes);
    nms_kernel<<<1, 1024, 0, stream>>>(sortedBoxes, keepPos, keptCount);

    const int ntot = PAD * 4 + PAD * CLASSES;
    output_kernel<<<(ntot + 255) / 256, 256, 0, stream>>>(
        sortedBoxes, scores, order, keepPos, keptCount, out);
}